// GaussianLayer_58050777972779
// MI455X (gfx1250) — compile-verified
//
#include <hip/hip_runtime.h>

typedef __attribute__((ext_vector_type(2))) float v2f;
typedef __attribute__((ext_vector_type(8))) float v8f;

#define KSIZE 13
#define PAD 6
#define IMG 1024
#define TILE 64               // 64x64 output tile per workgroup
#define INW (TILE + 2*PAD)    // 76: input region width (cols)
#define INROWS 80             // input rows padded to 5 wmma row-tiles
#define MIDW 74               // LDS stride for mid: conflict-free for both passes
#define NTHREADS 256

__global__ __launch_bounds__(NTHREADS)
void gauss13_sep_wmma(const float* __restrict__ x,
                      const float* __restrict__ w,
                      float* __restrict__ out)
{
    __shared__ float s_in[INROWS * INW];    // rows 0..75 = reflect-padded input, 76..79 = zero
    __shared__ float s_mid[INROWS * MIDW];  // horizontal-pass result (rows 0..79)
    __shared__ float s_k[16];               // 1D gaussian taps (13 used)

    const int tid = threadIdx.x;
    const int img = blockIdx.z;             // n*8 + c, 0..63
    const int ch  = img & 7;
    const int x0  = blockIdx.x * TILE;
    const int y0  = blockIdx.y * TILE;

    // ---- recover separable 1D kernel: k1[i] = k2[i][6] / sqrt(k2[6][6])
    if (tid < 16) s_k[tid] = 0.0f;
    if (tid < KSIZE) {
        const float* kc = w + ch * (KSIZE * KSIZE);
        float center = kc[6 * KSIZE + 6];
        s_k[tid] = kc[tid * KSIZE + 6] * rsqrtf(center);
    }

    // ---- cooperative load of 76x76 region with reflection padding
    const float* xin = x + (size_t)img * IMG * IMG;
    for (int idx = tid; idx < INW * INW; idx += NTHREADS) {
        int rr = idx / INW;
        int cc = idx - rr * INW;
        int gy = y0 - PAD + rr;
        int gx = x0 - PAD + cc;
        gy = gy < 0 ? -gy : gy;  gy = gy >= IMG ? 2 * IMG - 2 - gy : gy;
        gx = gx < 0 ? -gx : gx;  gx = gx >= IMG ? 2 * IMG - 2 - gx : gx;
        s_in[idx] = xin[(size_t)gy * IMG + gx];
    }
    // zero-fill pad rows 76..79 (their mid results are computed but never read)
    for (int idx = tid; idx < (INROWS - INW) * INW; idx += NTHREADS)
        s_in[INW * INW + idx] = 0.0f;
    __syncthreads();

    // ---- per-wave band operand: value k1[(4kk+kb) - (lane%16)], chunked K=4.
    // Serves as B (weights) in the horizontal pass and as A (weights) in the
    // vertical pass: both are band matrices k1[Kindex - laneIndex].
    const int lane  = tid & 31;
    const int swave = __builtin_amdgcn_readfirstlane(tid >> 5); // scalar wave id 0..7
    const int iA    = lane & 15;            // A-row / B-col / D-col index
    const int kb    = (lane >> 4) << 1;     // K-pair base: 0 (lanes 0-15) or 2 (lanes 16-31)

    v2f aband[7];
    #pragma unroll
    for (int kk = 0; kk < 7; ++kk) {
        int d0 = 4 * kk + kb - iA;
        int d1 = d0 + 1;
        aband[kk].x = ((unsigned)d0 < 13u) ? s_k[d0] : 0.0f;
        aband[kk].y = ((unsigned)d1 < 13u) ? s_k[d1] : 0.0f;
    }

    // ---- horizontal pass via WMMA: mid(16x16) = In(16x28) x W_band(28x16)
    // 20 tiles over 8 waves; scalar (SGPR) loop -> EXEC stays all-1s at WMMA
    for (int t = swave; t < 20; t += 8) {
        int ty = t >> 2, tx = t & 3;
        int row0 = ty * 16, c0 = tx * 16;

        v8f acc = {};
        #pragma unroll
        for (int kk = 0; kk < 7; ++kk) {
            // A: contiguous pair s_in[row0+iA][c0 + 4kk + kb .. +1] -> ds_load_b64
            v2f a = *(const v2f*)&s_in[(row0 + iA) * INW + c0 + 4 * kk + kb];
            acc = __builtin_amdgcn_wmma_f32_16x16x4_f32(
                false, a, false, aband[kk], (short)0, acc, false, false);
        }
        // D element v -> mid row (row0 + v + 8*(lane>>4)), col (c0 + iA)
        int mrow = row0 + ((lane >> 4) << 3);
        #pragma unroll
        for (int v = 0; v < 8; ++v)
            s_mid[(mrow + v) * MIDW + c0 + iA] = acc[v];
    }
    __syncthreads();

    // ---- vertical pass via WMMA: D(16x16) = A_band(16x28) x Mid(28x16)
    float* outp = out + (size_t)img * IMG * IMG;

    #pragma unroll
    for (int rep = 0; rep < 2; ++rep) {
        int tile = swave + rep * 8;         // 16 tiles of 16x16 over 8 waves
        int ty = tile >> 2, tx = tile & 3;
        int row0 = ty * 16, col0 = tx * 16;

        v8f acc = {};
        #pragma unroll
        for (int kk = 0; kk < 7; ++kk) {
            int p = 4 * kk + kb;
            const float* bp = &s_mid[(row0 + p) * MIDW + col0 + iA];
            v2f b;
            b.x = bp[0];
            b.y = bp[MIDW];
            acc = __builtin_amdgcn_wmma_f32_16x16x4_f32(
                false, aband[kk], false, b, (short)0, acc, false, false);
        }

        // D element v -> out row (row0 + v + 8*(lane>>4)), col (col0 + iA)
        int orow = y0 + row0 + ((lane >> 4) << 3);
        int ocol = x0 + col0 + iA;
        #pragma unroll
        for (int v = 0; v < 8; ++v)
            outp[(size_t)(orow + v) * IMG + ocol] = acc[v];
    }
}

extern "C" void kernel_launch(void* const* d_in, const int* in_sizes, int n_in,
                              void* d_out, int out_size, void* d_ws, size_t ws_size,
                              hipStream_t stream) {
    const float* x = (const float*)d_in[0];   // (8, 8, 1024, 1024) fp32
    const float* w = (const float*)d_in[1];   // (8, 1, 13, 13) fp32
    float* o = (float*)d_out;                 // (8, 8, 1024, 1024) fp32
    dim3 grid(IMG / TILE, IMG / TILE, 64);    // 16 x 16 x (8 batches * 8 channels)
    gauss13_sep_wmma<<<grid, NTHREADS, 0, stream>>>(x, w, o);
}